// NeuralMemoryModule_39951785788101
// MI455X (gfx1250) — compile-verified
//
#include <hip/hip_runtime.h>
#include <math.h>

typedef float v2f __attribute__((ext_vector_type(2)));
typedef float v8f __attribute__((ext_vector_type(8)));

#define BB 32
#define SS 2048
#define DD 1024
#define MM 1024
#define CH 8
#define BM (BB * MM)
#define BD (BB * DD)

// ---------------------------------------------------------------------------
// Block-wide deterministic sum over 256 threads (8 waves, wave32).
// ---------------------------------------------------------------------------
__device__ __forceinline__ float block_sum256(float v, float* sbuf) {
#pragma unroll
  for (int o = 16; o > 0; o >>= 1) v += __shfl_down(v, o, 32);
  __syncthreads();
  if ((threadIdx.x & 31) == 0) sbuf[threadIdx.x >> 5] = v;
  __syncthreads();
  float t = 0.f;
#pragma unroll
  for (int w = 0; w < 8; ++w) t += sbuf[w];
  return t;
}

// ---------------------------------------------------------------------------
// WMMA fp32 GEMM: C[r,n] = sum_k A[r,k] * B[k,n]   (or B[n,k] if transB)
// Rows = 32 (two 16-row tiles). One wave per 16x16 output tile.
// Optional: bias[n] (scaled by rowScale[r] if given), sigmoid epilogue.
// Uses V_WMMA_F32_16X16X4_F32 (full fp32 precision, wave32).
// ---------------------------------------------------------------------------
__global__ void __launch_bounds__(32)
wmma_gemm_k(const float* __restrict__ A, int lda,
            const float* __restrict__ Bm, int ldb, int transB,
            const float* __restrict__ bias, const float* __restrict__ rowScale,
            float* __restrict__ C, int ldc, int K, int act) {
  const int lane = threadIdx.x;
  const int r0 = blockIdx.y << 4;
  const int n0 = blockIdx.x << 4;
  const int rl = lane & 15;
  const int kh = (lane >> 4) << 1;  // lanes 0-15: K pair {0,1}; lanes 16-31: {2,3}
  const float* Arow = A + (long)(r0 + rl) * lda;
  v8f acc = {};
  if (!transB) {
    const float* Bc = Bm + (n0 + rl);
    for (int k = 0; k < K; k += 4) {
      v2f a, b;
      a.x = Arow[k + kh];
      a.y = Arow[k + kh + 1];
      b.x = Bc[(long)(k + kh) * ldb];
      b.y = Bc[(long)(k + kh + 1) * ldb];
      acc = __builtin_amdgcn_wmma_f32_16x16x4_f32(false, a, false, b,
                                                  (short)0, acc, false, false);
    }
  } else {
    const float* Bc = Bm + (long)(n0 + rl) * ldb;
    for (int k = 0; k < K; k += 4) {
      v2f a, b;
      a.x = Arow[k + kh];
      a.y = Arow[k + kh + 1];
      b.x = Bc[k + kh];
      b.y = Bc[k + kh + 1];
      acc = __builtin_amdgcn_wmma_f32_16x16x4_f32(false, a, false, b,
                                                  (short)0, acc, false, false);
    }
  }
  const int col = n0 + rl;
  const float bcol = bias ? bias[col] : 0.0f;
  const int rbase = r0 + ((lane >> 4) << 3);  // C: VGPR v holds rows v and v+8
#pragma unroll
  for (int v = 0; v < 8; ++v) {
    const int rr = rbase + v;
    float val = acc[v];
    val += rowScale ? bcol * rowScale[rr] : bcol;
    if (act == 1) val = 1.0f / (1.0f + expf(-val));
    C[(long)rr * ldc + col] = val;
  }
}

// ---------------------------------------------------------------------------
// LayerNorm (eps=1e-5) + affine + SiLU, one block per row, saves activations.
// ---------------------------------------------------------------------------
__global__ void __launch_bounds__(256)
ln_silu_k(const float* __restrict__ z, const float* __restrict__ gamma,
          const float* __restrict__ beta, float* __restrict__ xhat,
          float* __restrict__ ys, float* __restrict__ rstd,
          float* __restrict__ hout, int Mn) {
  __shared__ float sbuf[8];
  const int b = blockIdx.x;
  const long base = (long)b * Mn;
  const float* zr = z + base;
  float s = 0.f;
  for (int i = threadIdx.x; i < Mn; i += 256) s += zr[i];
  const float mu = block_sum256(s, sbuf) / (float)Mn;
  float v = 0.f;
  for (int i = threadIdx.x; i < Mn; i += 256) {
    float d = zr[i] - mu;
    v += d * d;
  }
  const float var = block_sum256(v, sbuf) / (float)Mn;
  const float rs = rsqrtf(var + 1e-5f);
  if (threadIdx.x == 0) rstd[b] = rs;
  for (int i = threadIdx.x; i < Mn; i += 256) {
    float xh = (zr[i] - mu) * rs;
    xhat[base + i] = xh;
    float y = xh * gamma[i] + beta[i];
    ys[base + i] = y;
    float sg = 1.0f / (1.0f + expf(-y));
    hout[base + i] = y * sg;
  }
}

// ---------------------------------------------------------------------------
// Backward through SiLU + affine + LayerNorm: grad wrt pre-LN z.
// ---------------------------------------------------------------------------
__global__ void __launch_bounds__(256)
ln_silu_bwd_k(const float* __restrict__ gh, const float* __restrict__ ys,
              const float* __restrict__ xhat, const float* __restrict__ rstd,
              const float* __restrict__ gamma, float* __restrict__ gz, int Mn) {
  __shared__ float sbuf[8];
  const int b = blockIdx.x;
  const long base = (long)b * Mn;
  float s1 = 0.f, s2 = 0.f;
  for (int i = threadIdx.x; i < Mn; i += 256) {
    float y = ys[base + i];
    float sg = 1.0f / (1.0f + expf(-y));
    float gy = gh[base + i] * sg * (1.0f + y * (1.0f - sg));  // dSiLU
    float gx = gy * gamma[i];
    s1 += gx;
    s2 += gx * xhat[base + i];
  }
  const float t1 = block_sum256(s1, sbuf) / (float)Mn;
  const float t2 = block_sum256(s2, sbuf) / (float)Mn;
  const float rs = rstd[b];
  for (int i = threadIdx.x; i < Mn; i += 256) {
    float y = ys[base + i];
    float sg = 1.0f / (1.0f + expf(-y));
    float gy = gh[base + i] * sg * (1.0f + y * (1.0f - sg));
    float gx = gy * gamma[i];
    gz[base + i] = rs * (gx - t1 - xhat[base + i] * t2);
  }
}

// ---------------------------------------------------------------------------
// cb[b] = bk . mo[b]
// ---------------------------------------------------------------------------
__global__ void __launch_bounds__(256)
dotrow_k(const float* __restrict__ a, const float* __restrict__ Bv,
         float* __restrict__ outv, int Mn) {
  __shared__ float sbuf[8];
  const int b = blockIdx.x;
  float s = 0.f;
  for (int i = threadIdx.x; i < Mn; i += 256) s += a[i] * Bv[(long)b * Mn + i];
  const float t = block_sum256(s, sbuf);
  if (threadIdx.x == 0) outv[b] = t;
}

// ---------------------------------------------------------------------------
// wv_sum[d] = sum_m Wv[d,m]; last block reduces bv into bv_sum.
// ---------------------------------------------------------------------------
__global__ void __launch_bounds__(256)
wvsum_k(const float* __restrict__ Wv, const float* __restrict__ bv,
        float* __restrict__ wvs, float* __restrict__ bvs, int Mn, int Dn) {
  __shared__ float sbuf[8];
  const int d = blockIdx.x;
  if (d < Dn) {
    float s = 0.f;
    for (int i = threadIdx.x; i < Mn; i += 256) s += Wv[(long)d * Mn + i];
    const float t = block_sum256(s, sbuf);
    if (threadIdx.x == 0) wvs[d] = t;
  } else {
    float s = 0.f;
    for (int i = threadIdx.x; i < Mn; i += 256) s += bv[i];
    const float t = block_sum256(s, sbuf);
    if (threadIdx.x == 0) bvs[0] = t;
  }
}

// ---------------------------------------------------------------------------
// The bandwidth-bound sweep over inputs [B,S,D] (256 MB, read once).
// Per row s: p = x.wk_mo[b], v = x.wv_sum; g = (M*(p+cb)-(v+bvsum))/(B*S*M);
// accumulate u += g*x, pooled += x, G += g. One block per (b, S-chunk).
// Deterministic: wave-serialized LDS accumulation, 2-pass global reduce.
// ---------------------------------------------------------------------------
__global__ void __launch_bounds__(256)
sweep_k(const float* __restrict__ x, const float* __restrict__ wkmo,
        const float* __restrict__ wvs, const float* __restrict__ cb,
        const float* __restrict__ bvs, float* __restrict__ pu,
        float* __restrict__ pp, float* __restrict__ pG) {
  __shared__ float accu[DD];
  __shared__ float accp[DD];
  __shared__ float red8[8];
  const int b = blockIdx.x / CH;
  const int ch = blockIdx.x % CH;
  const int lane = threadIdx.x & 31;
  const int wid = threadIdx.x >> 5;

  const float4* wk4 = (const float4*)(wkmo + (size_t)b * DD);
  const float4* wv4 = (const float4*)wvs;
  float4 wkr[8], wvr[8], uac[8], pac[8];
#pragma unroll
  for (int i = 0; i < 8; ++i) {
    wkr[i] = wk4[i * 32 + lane];
    wvr[i] = wv4[i * 32 + lane];
    uac[i] = make_float4(0.f, 0.f, 0.f, 0.f);
    pac[i] = make_float4(0.f, 0.f, 0.f, 0.f);
  }
  const float cbb = cb[b];
  const float bvsum = bvs[0];
  const float inv = 1.0f / ((float)BB * (float)SS * (float)MM);
  float Gacc = 0.f;
  const int ROWS = SS / CH;  // 256 rows per chunk, 32 rows per wave

  for (int j = 0; j < ROWS / 8; ++j) {
    const int s = ch * ROWS + j * 8 + wid;
    const float4* xr = (const float4*)(x + ((size_t)b * SS + s) * DD);
    float4 xv[8];
    float p = 0.f, vv = 0.f;
#pragma unroll
    for (int i = 0; i < 8; ++i) {
      xv[i] = xr[i * 32 + lane];
      p = fmaf(xv[i].x, wkr[i].x, p); p = fmaf(xv[i].y, wkr[i].y, p);
      p = fmaf(xv[i].z, wkr[i].z, p); p = fmaf(xv[i].w, wkr[i].w, p);
      vv = fmaf(xv[i].x, wvr[i].x, vv); vv = fmaf(xv[i].y, wvr[i].y, vv);
      vv = fmaf(xv[i].z, wvr[i].z, vv); vv = fmaf(xv[i].w, wvr[i].w, vv);
    }
#pragma unroll
    for (int o2 = 16; o2 > 0; o2 >>= 1) {
      p += __shfl_xor(p, o2, 32);
      vv += __shfl_xor(vv, o2, 32);
    }
    const float g = ((float)MM * (p + cbb) - (vv + bvsum)) * inv;
    Gacc += g;
#pragma unroll
    for (int i = 0; i < 8; ++i) {
      uac[i].x = fmaf(g, xv[i].x, uac[i].x);
      uac[i].y = fmaf(g, xv[i].y, uac[i].y);
      uac[i].z = fmaf(g, xv[i].z, uac[i].z);
      uac[i].w = fmaf(g, xv[i].w, uac[i].w);
      pac[i].x += xv[i].x; pac[i].y += xv[i].y;
      pac[i].z += xv[i].z; pac[i].w += xv[i].w;
    }
  }

  for (int i = threadIdx.x; i < DD; i += 256) { accu[i] = 0.f; accp[i] = 0.f; }
  __syncthreads();
  for (int w8 = 0; w8 < 8; ++w8) {  // fixed order across waves -> deterministic
    if (wid == w8) {
#pragma unroll
      for (int i = 0; i < 8; ++i) {
        const int d = i * 128 + lane * 4;
        accu[d + 0] += uac[i].x; accu[d + 1] += uac[i].y;
        accu[d + 2] += uac[i].z; accu[d + 3] += uac[i].w;
        accp[d + 0] += pac[i].x; accp[d + 1] += pac[i].y;
        accp[d + 2] += pac[i].z; accp[d + 3] += pac[i].w;
      }
    }
    __syncthreads();
  }
  const size_t base = ((size_t)b * CH + ch) * DD;
  for (int i = threadIdx.x; i < DD; i += 256) {
    pu[base + i] = accu[i];
    pp[base + i] = accp[i];
  }
  if (lane == 0) red8[wid] = Gacc;  // g uniform across lanes after xor-reduce
  __syncthreads();
  if (threadIdx.x == 0) {
    float t = 0.f;
#pragma unroll
    for (int i2 = 0; i2 < 8; ++i2) t += red8[i2];
    pG[b * CH + ch] = t;
  }
}

// ---------------------------------------------------------------------------
// Reduce chunk partials; also build gate_in = [pooled, memory_state].
// ---------------------------------------------------------------------------
__global__ void __launch_bounds__(256)
reduce_k(const float* __restrict__ pu, const float* __restrict__ pp,
         const float* __restrict__ pG, const float* __restrict__ mem,
         float* __restrict__ u, float* __restrict__ gatein,
         float* __restrict__ G) {
  const int b = blockIdx.x;
  for (int d = threadIdx.x; d < DD; d += 256) {
    float su = 0.f, sp = 0.f;
    for (int c = 0; c < CH; ++c) {
      const size_t idx = ((size_t)b * CH + c) * DD + d;
      su += pu[idx];
      sp += pp[idx];
    }
    u[(size_t)b * DD + d] = su;
    gatein[(size_t)b * (DD + MM) + d] = sp * (1.0f / (float)SS);
    gatein[(size_t)b * (DD + MM) + DD + d] = mem[(size_t)b * MM + d];
  }
  if (threadIdx.x == 0) {
    float t = 0.f;
    for (int c = 0; c < CH; ++c) t += pG[b * CH + c];
    G[b] = t;
  }
}

// ---------------------------------------------------------------------------
// new_momentum = eta*mom + theta*grad; new_mem = (1-fg)*mem + ug*new_momentum
// ---------------------------------------------------------------------------
__global__ void __launch_bounds__(256)
update_k(const float* __restrict__ mem, const float* __restrict__ mom,
         const float* __restrict__ sg, const float* __restrict__ fg,
         const float* __restrict__ ug, const float* __restrict__ eta,
         const float* __restrict__ theta, float* __restrict__ nmem,
         float* __restrict__ out2, int n) {
  const int i = blockIdx.x * 256 + threadIdx.x;
  if (i >= n) return;
  const float nm = eta[0] * mom[i] + theta[0] * sg[i];
  const float v = (1.0f - fg[i]) * mem[i] + ug[i] * nm;
  nmem[i] = v;
  out2[i] = v;
}

// ---------------------------------------------------------------------------
extern "C" void kernel_launch(void* const* d_in, const int* in_sizes, int n_in,
                              void* d_out, int out_size, void* d_ws,
                              size_t ws_size, hipStream_t stream) {
  (void)in_sizes; (void)n_in; (void)out_size; (void)ws_size;
  const float* inp   = (const float*)d_in[0];
  const float* mem0  = (const float*)d_in[1];
  const float* mom0  = (const float*)d_in[2];
  const float* Wk    = (const float*)d_in[3];
  const float* bk    = (const float*)d_in[4];
  const float* Wv    = (const float*)d_in[5];
  const float* bv    = (const float*)d_in[6];
  const float* memW  = (const float*)d_in[7];
  const float* memb  = (const float*)d_in[8];
  const float* lng   = (const float*)d_in[9];
  const float* lnb   = (const float*)d_in[10];
  const float* Wf    = (const float*)d_in[11];
  const float* bf    = (const float*)d_in[12];
  const float* Wu    = (const float*)d_in[15];  // Wm/bm unused in reference
  const float* bu    = (const float*)d_in[16];
  const float* eta   = (const float*)d_in[17];
  const float* theta = (const float*)d_in[18];
  float* out = (float*)d_out;

  float* w = (float*)d_ws;
  size_t o = 0;
  float* z0 = w + o; o += BM;  float* z1 = w + o; o += BM;
  float* xh0 = w + o; o += BM; float* xh1 = w + o; o += BM;
  float* y0 = w + o; o += BM;  float* y1 = w + o; o += BM;
  float* h1 = w + o; o += BM;  float* mo = w + o; o += BM;
  float* rs0 = w + o; o += BB; float* rs1 = w + o; o += BB;
  float* wkmo = w + o; o += BD;
  float* cb = w + o; o += BB;
  float* wvs = w + o; o += DD;
  float* bvs = w + o; o += 16;
  float* pu = w + o; o += (size_t)BB * CH * DD;
  float* pp = w + o; o += (size_t)BB * CH * DD;
  float* pG = w + o; o += BB * CH;
  float* u = w + o; o += BD;
  float* gin = w + o; o += (size_t)BB * (DD + MM);
  float* G = w + o; o += BB;
  float* ga = w + o; o += BM;  float* gb = w + o; o += BM;
  float* fg = w + o; o += BM;  float* ug = w + o; o += BM;
  float* nmem = w + o; o += BM;

  const dim3 g16(MM / 16, BB / 16);  // 64 x 2 tiles
  const dim3 gD(DD / 16, BB / 16);
  const dim3 b32(32), b256(256);

  // Phase A: mo = deep-memory MLP(memory_state), saving activations.
  wmma_gemm_k<<<g16, b32, 0, stream>>>(mem0, MM, memW, MM, 0, memb, nullptr, z0, MM, MM, 0);
  ln_silu_k<<<BB, b256, 0, stream>>>(z0, lng, lnb, xh0, y0, rs0, h1, MM);
  wmma_gemm_k<<<g16, b32, 0, stream>>>(h1, MM, memW + MM * MM, MM, 0, memb + MM, nullptr, z1, MM, MM, 0);
  ln_silu_k<<<BB, b256, 0, stream>>>(z1, lng + MM, lnb + MM, xh1, y1, rs1, mo, MM);

  // Phase B: collapse keys/values contractions: wk_mo = mo.Wk^T, cb, wv_sum.
  wmma_gemm_k<<<gD, b32, 0, stream>>>(mo, MM, Wk, MM, 1, nullptr, nullptr, wkmo, DD, MM, 0);
  dotrow_k<<<BB, b256, 0, stream>>>(bk, mo, cb, MM);
  wvsum_k<<<DD + 1, b256, 0, stream>>>(Wv, bv, wvs, bvs, MM, DD);

  // Phase C: one streaming pass over inputs (256 MB) -> u, pooled, G.
  sweep_k<<<BB * CH, b256, 0, stream>>>(inp, wkmo, wvs, cb, bvs, pu, pp, pG);
  reduce_k<<<BB, b256, 0, stream>>>(pu, pp, pG, mem0, u, gin, G);

  // Phase D: surprise_grad = MLP backward of (u.Wk + G*bk).
  wmma_gemm_k<<<g16, b32, 0, stream>>>(u, DD, Wk, MM, 0, bk, G, ga, MM, DD, 0);
  ln_silu_bwd_k<<<BB, b256, 0, stream>>>(ga, y1, xh1, rs1, lng + MM, gb, MM);
  wmma_gemm_k<<<g16, b32, 0, stream>>>(gb, MM, memW + MM * MM, MM, 1, nullptr, nullptr, ga, MM, MM, 0);
  ln_silu_bwd_k<<<BB, b256, 0, stream>>>(ga, y0, xh0, rs0, lng, gb, MM);
  wmma_gemm_k<<<g16, b32, 0, stream>>>(gb, MM, memW, MM, 1, nullptr, nullptr, ga, MM, MM, 0);
  // ga = surprise_grad [B,M]

  // Phase E: gates (momentum gate skipped; unused in reference update).
  wmma_gemm_k<<<g16, b32, 0, stream>>>(gin, DD + MM, Wf, MM, 0, bf, nullptr, fg, MM, DD + MM, 1);
  wmma_gemm_k<<<g16, b32, 0, stream>>>(gin, DD + MM, Wu, MM, 0, bu, nullptr, ug, MM, DD + MM, 1);

  // Phase F: memory update -> ws + second half of d_out.
  update_k<<<BM / 256, b256, 0, stream>>>(mem0, mom0, ga, fg, ug, eta, theta, nmem, out + BM, BM);

  // Phase G: processed = MLP(new_memory) -> first half of d_out.
  wmma_gemm_k<<<g16, b32, 0, stream>>>(nmem, MM, memW, MM, 0, memb, nullptr, z0, MM, MM, 0);
  ln_silu_k<<<BB, b256, 0, stream>>>(z0, lng, lnb, xh0, y0, rs0, h1, MM);
  wmma_gemm_k<<<g16, b32, 0, stream>>>(h1, MM, memW + MM * MM, MM, 0, memb + MM, nullptr, z1, MM, MM, 0);
  ln_silu_k<<<BB, b256, 0, stream>>>(z1, lng + MM, lnb + MM, xh1, y1, rs1, out, MM);
}